// AggressiveLoss_71012989272121
// MI455X (gfx1250) — compile-verified
//
#include <hip/hip_runtime.h>

// Problem constants (from reference): B=4096, C=10, H=W=30
#define NB   4096
#define NC   10
#define NHW  900        // 30*30 pixels per sample
#define NG   225        // pixel-quads per sample (900/4), one float4 per channel per quad
#define NPIX_TOTAL 3686400.0f   // B*H*W

typedef float v4f __attribute__((ext_vector_type(4)));

static __device__ __forceinline__ v4f nt_load4(const float* p) {
    return __builtin_nontemporal_load((const v4f*)p);
}

// ---------------------------------------------------------------------------
// Stage 1: one block per sample b. 256 threads (8 wave32); threads 0..224 each
// own a quad of 4 consecutive pixels. Per-channel loads are 128-bit, fully
// coalesced, non-temporal (stream-once, dataset >> 192MB L2).
// Emits per-sample partials to ws[b*8 + {0..4}]:
//   0: sum over pixels of ce * (1 + 2*incorrect)
//   1: exact (all pred_idx == target_idx)
//   2: should_not_copy * did_copy
//   3: (changed - t_changed)^2
//   4: popcount(target colors missing from prediction)
// ---------------------------------------------------------------------------
__global__ __launch_bounds__(256) void loss_stage1(
    const float* __restrict__ pred,
    const float* __restrict__ targ,
    const float* __restrict__ inp,
    float* __restrict__ ws)
{
    const int b = blockIdx.x;
    const int t = threadIdx.x;

    float ce_w = 0.0f;   // weighted CE accumulator
    int   cnt  = 0;      // bits[9:0]=#(p==t), bits[19:10]=#(p==i), bits[29:20]=#(t==i)
    int   msk  = 0;      // bits[9:0]=pred colors present, bits[19:10]=target colors present

    if (t < NG) {
        const size_t base = (size_t)b * (NC * NHW) + (size_t)t * 4;

        // ---- load all 10 pred channels for this quad (stays in VGPRs) ----
        v4f p[NC];
        #pragma unroll
        for (int c = 0; c < NC; ++c)
            p[c] = nt_load4(pred + base + c * NHW);

        // ---- target pass: argmax + the pred value at the target channel ----
        int   tidx[4];
        float tpred[4];
        {
            v4f tv = nt_load4(targ + base);
            float tbest[4];
            #pragma unroll
            for (int j = 0; j < 4; ++j) { tbest[j] = tv[j]; tidx[j] = 0; tpred[j] = p[0][j]; }
            #pragma unroll
            for (int c = 1; c < NC; ++c) {
                v4f cv = nt_load4(targ + base + c * NHW);
                #pragma unroll
                for (int j = 0; j < 4; ++j) {
                    if (cv[j] > tbest[j]) { tbest[j] = cv[j]; tidx[j] = c; tpred[j] = p[c][j]; }
                }
            }
        }

        // ---- input pass: argmax only ----
        int iidx[4];
        {
            v4f iv = nt_load4(inp + base);
            float ibest[4];
            #pragma unroll
            for (int j = 0; j < 4; ++j) { ibest[j] = iv[j]; iidx[j] = 0; }
            #pragma unroll
            for (int c = 1; c < NC; ++c) {
                v4f cv = nt_load4(inp + base + c * NHW);
                #pragma unroll
                for (int j = 0; j < 4; ++j) {
                    if (cv[j] > ibest[j]) { ibest[j] = cv[j]; iidx[j] = c; }
                }
            }
        }

        // ---- per-pixel: pred argmax, log-sum-exp, flags ----
        #pragma unroll
        for (int j = 0; j < 4; ++j) {
            float pm = p[0][j]; int pidx = 0;
            #pragma unroll
            for (int c = 1; c < NC; ++c) {
                if (p[c][j] > pm) { pm = p[c][j]; pidx = c; }
            }
            float se = 0.0f;
            #pragma unroll
            for (int c = 0; c < NC; ++c)
                se += __expf(p[c][j] - pm);
            // accurate log for the small correction term
            float ce = pm + logf(se) - tpred[j];

            const bool corr = (pidx == tidx[j]);
            ce_w += ce * (corr ? 1.0f : 3.0f);
            cnt  += (corr ? 1 : 0)
                  + ((pidx == iidx[j])    ? (1 << 10) : 0)
                  + ((tidx[j] == iidx[j]) ? (1 << 20) : 0);
            msk  |= (1 << pidx) | (1 << (10 + tidx[j]));
        }
    }

    // ---- wave32 reduction ----
    #pragma unroll
    for (int off = 16; off > 0; off >>= 1) {
        ce_w += __shfl_down(ce_w, off, 32);
        cnt  += __shfl_down(cnt,  off, 32);
        msk  |= __shfl_down(msk,  off, 32);
    }

    // ---- cross-wave (8 waves) via LDS ----
    __shared__ float s_ce[8];
    __shared__ int   s_cnt[8];
    __shared__ int   s_msk[8];
    const int wave = t >> 5;
    const int lane = t & 31;
    if (lane == 0) { s_ce[wave] = ce_w; s_cnt[wave] = cnt; s_msk[wave] = msk; }
    __syncthreads();

    if (t == 0) {
        float ceS = 0.0f; int cS = 0, mS = 0;
        #pragma unroll
        for (int w = 0; w < 8; ++w) { ceS += s_ce[w]; cS += s_cnt[w]; mS |= s_msk[w]; }

        const int cpt = cS & 1023;          // pixels with pred==target
        const int cpi = (cS >> 10) & 1023;  // pixels with pred==input
        const int cti = (cS >> 20) & 1023;  // pixels with target==input

        const float exact      = (cpt == NHW) ? 1.0f : 0.0f;
        const float did_copy   = (cpi == NHW) ? 1.0f : 0.0f;
        const float should_not = (cti != NHW) ? 1.0f : 0.0f;
        const float changed    = (float)(NHW - cpi) * (1.0f / (float)NHW);
        const float t_changed  = (float)(NHW - cti) * (1.0f / (float)NHW);
        const int   pmask      = mS & 1023;
        const int   tmask      = (mS >> 10) & 1023;
        const float missing    = (float)__popc(tmask & ~pmask);

        float* o = ws + (size_t)b * 8;
        o[0] = ceS;
        o[1] = exact;
        o[2] = should_not * did_copy;
        o[3] = (changed - t_changed) * (changed - t_changed);
        o[4] = missing;
    }
}

// ---------------------------------------------------------------------------
// Stage 2: single 256-thread block, fixed-order reduction over 4096 samples.
// d_out layout (fp32): [total, ce_loss, exact_bonus, copy_penalty,
//                       transform_diff, sum(exact)]
// ---------------------------------------------------------------------------
__global__ __launch_bounds__(256) void loss_stage2(
    const float* __restrict__ ws,
    float* __restrict__ out)
{
    const int t = threadIdx.x;
    float s0 = 0.f, s1 = 0.f, s2 = 0.f, s3 = 0.f, s4 = 0.f;
    for (int b = t; b < NB; b += 256) {
        const float* o = ws + (size_t)b * 8;
        s0 += o[0]; s1 += o[1]; s2 += o[2]; s3 += o[3]; s4 += o[4];
    }

    #pragma unroll
    for (int off = 16; off > 0; off >>= 1) {
        s0 += __shfl_down(s0, off, 32);
        s1 += __shfl_down(s1, off, 32);
        s2 += __shfl_down(s2, off, 32);
        s3 += __shfl_down(s3, off, 32);
        s4 += __shfl_down(s4, off, 32);
    }

    __shared__ float red[8][5];
    const int wave = t >> 5;
    const int lane = t & 31;
    if (lane == 0) {
        red[wave][0] = s0; red[wave][1] = s1; red[wave][2] = s2;
        red[wave][3] = s3; red[wave][4] = s4;
    }
    __syncthreads();

    if (t == 0) {
        float a0 = 0.f, a1 = 0.f, a2 = 0.f, a3 = 0.f, a4 = 0.f;
        #pragma unroll
        for (int w = 0; w < 8; ++w) {
            a0 += red[w][0]; a1 += red[w][1]; a2 += red[w][2];
            a3 += red[w][3]; a4 += red[w][4];
        }
        const float invB = 1.0f / (float)NB;
        const float ce_mean     = a0 * (1.0f / NPIX_TOTAL);
        const float ce_loss     = ce_mean + 0.5f * a4;
        const float exact_bonus = -10.0f * a1 * invB;
        const float copy_pen    = 5.0f * a2 * invB;
        const float tdiff       = 2.0f * a3 * invB;
        out[0] = ce_loss + exact_bonus + copy_pen + tdiff;
        out[1] = ce_loss;
        out[2] = exact_bonus;
        out[3] = copy_pen;
        out[4] = tdiff;
        out[5] = a1;
    }
}

extern "C" void kernel_launch(void* const* d_in, const int* in_sizes, int n_in,
                              void* d_out, int out_size, void* d_ws, size_t ws_size,
                              hipStream_t stream) {
    (void)in_sizes; (void)n_in; (void)out_size; (void)ws_size;
    const float* pred = (const float*)d_in[0];
    const float* targ = (const float*)d_in[1];
    const float* inp  = (const float*)d_in[2];
    float* ws  = (float*)d_ws;    // needs NB*8*4 = 128 KB
    float* out = (float*)d_out;   // 6 fp32 scalars

    loss_stage1<<<NB, 256, 0, stream>>>(pred, targ, inp, ws);
    loss_stage2<<<1, 256, 0, stream>>>(ws, out);
}